// TripletLoss_13331578487552
// MI455X (gfx1250) — compile-verified
//
#include <hip/hip_runtime.h>

// ---------------------------------------------------------------------------
// Triplet loss (batch-semi-hard) fused kernel for MI455X / gfx1250.
// d2 = sq_i + sq_j - 2*E@E^T computed tile-wise with V_WMMA_F32_16X16X4_F32
// (fp32-exact GEMM; mining uses strict margin comparisons, so no f16).
// E = 2MB -> fully L2 resident; the 256MB B x B matrix is never materialized.
// Hot loop = 16 WMMA + 16 b64 loads + branchless select/min epilogue.
// ---------------------------------------------------------------------------

typedef __attribute__((ext_vector_type(2))) float v2f;
typedef __attribute__((ext_vector_type(8))) float v8f;

#define TL_MARGIN 0.2f
#define TL_BIG    1e30f
#define TL_D      64
#define TL_NK     16   // D / 4 K-chunks per WMMA tile
#define TL_INTMAX 0x7FFFFFFF

// ---- Prologue: per-row squared norms (B floats into workspace) ----
__global__ void sqnorm_kernel(const float* __restrict__ E,
                              float* __restrict__ sq, int B) {
  const int i = blockIdx.x * blockDim.x + threadIdx.x;
  if (i >= B) return;
  const float* row = E + (size_t)i * TL_D;
  float s = 0.f;
#pragma unroll
  for (int k = 0; k < TL_D; k += 4) {
    float4 v = *(const float4*)(row + k);
    s = fmaf(v.x, v.x, s);
    s = fmaf(v.y, v.y, s);
    s = fmaf(v.z, v.z, s);
    s = fmaf(v.w, v.w, s);
  }
  sq[i] = s;
}

__launch_bounds__(256)
__global__ void triplet_tiles_kernel(const float* __restrict__ E,
                                     const int*   __restrict__ labels,
                                     const float* __restrict__ sq,
                                     float*       __restrict__ partials, // [nRowTiles][2]
                                     int B) {
  const int nColTiles = B >> 4;
  const int tid  = threadIdx.x;
  const int wave = tid >> 5;
  const int lane = tid & 31;
  const int half = lane >> 4;    // 0: lanes 0..15, 1: lanes 16..31
  const int l16  = lane & 15;
  const int rowBase = blockIdx.x << 4;

  __shared__ int   s_lab[16];
  __shared__ float s_sqr[16];
  __shared__ float s_ap_w[8][16];
  __shared__ float s_ng_w[8][16];
  __shared__ int   s_cnt_w[8][16];
  __shared__ int   s_fj_w[8][16];
  __shared__ float s_ap[16];
  __shared__ float s_ng[16];
  __shared__ int   s_cnt[16];
  __shared__ int   s_fj[16];
  __shared__ int   s_rank[16];
  __shared__ float s_loss[16];
  __shared__ float s_val[16];

  if (tid < 16) {
    s_lab[tid]  = labels[rowBase + tid];
    s_sqr[tid]  = sq[rowBase + tid];
    s_rank[tid] = 0;
  }
  __syncthreads();

  // ---- A fragments: 16 rows x 64 K, fp32, per 16x4 f32 WMMA A layout.
  // lane l<16 holds row l, K = 4k+{0,1}; lane l>=16 holds row l-16, K = 4k+{2,3}.
  v2f afrag[TL_NK];
  {
    const float* arow = E + (size_t)(rowBase + l16) * TL_D + (half << 1);
#pragma unroll
    for (int k = 0; k < TL_NK; ++k)
      afrag[k] = *(const v2f*)(arow + (k << 2));
  }

  // C/D layout: lane<16 -> rows 0..7 (VGPR v = row v), lane>=16 -> rows 8..15.
  int rlab[8]; float rsq[8];
#pragma unroll
  for (int v = 0; v < 8; ++v) {
    rlab[v] = s_lab[half * 8 + v];
    rsq[v]  = s_sqr[half * 8 + v];
  }

  // =========================== PASS 1 ===========================
  // dist_ap (min same-class d2 > 0), class count, fallback min-negative.
  float apmin[8], ngmin[8];
  int   cnt[8];
#pragma unroll
  for (int v = 0; v < 8; ++v) { apmin[v] = TL_BIG; ngmin[v] = TL_BIG; cnt[v] = 0; }

  for (int t = wave; t < nColTiles; t += 8) {
    const int c = (t << 4) + l16;
    if (t + 8 < nColTiles)
      __builtin_prefetch(E + (size_t)(((t + 8) << 4) + l16) * TL_D, 0, 0);

    const float* brow = E + (size_t)c * TL_D + (half << 1);
    v2f bfrag[TL_NK];
#pragma unroll
    for (int k = 0; k < TL_NK; ++k)
      bfrag[k] = *(const v2f*)(brow + (k << 2));

    // two independent WMMA chains -> half the RAW-dependency depth
    v8f acc0 = {}, acc1 = {};
#pragma unroll
    for (int k = 0; k < TL_NK; k += 2) {
      acc0 = __builtin_amdgcn_wmma_f32_16x16x4_f32(false, afrag[k],
                                                   false, bfrag[k],
                                                   (short)0, acc0, false, false);
      acc1 = __builtin_amdgcn_wmma_f32_16x16x4_f32(false, afrag[k + 1],
                                                   false, bfrag[k + 1],
                                                   (short)0, acc1, false, false);
    }
    const float csq  = sq[c];
    const int   clab = labels[c];

#pragma unroll
    for (int v = 0; v < 8; ++v) {
      const float g  = acc0[v] + acc1[v];
      const float d2 = fmaxf(rsq[v] + csq - 2.0f * g, 0.0f);
      const bool same = (clab == rlab[v]);
      cnt[v]  += same ? 1 : 0;
      apmin[v] = fminf(apmin[v], (same && (d2 > 0.0f)) ? d2 : TL_BIG);
      ngmin[v] = fminf(ngmin[v], same ? TL_BIG : d2);
    }
  }

#pragma unroll
  for (int v = 0; v < 8; ++v) {
    float ap = apmin[v], ng = ngmin[v]; int cn = cnt[v];
    for (int m = 8; m >= 1; m >>= 1) {
      ap = fminf(ap, __shfl_xor(ap, m, 32));
      ng = fminf(ng, __shfl_xor(ng, m, 32));
      cn += __shfl_xor(cn, m, 32);
    }
    if (l16 == 0) {
      s_ap_w[wave][half * 8 + v]  = ap;
      s_ng_w[wave][half * 8 + v]  = ng;
      s_cnt_w[wave][half * 8 + v] = cn;
    }
  }
  __syncthreads();
  if (tid < 16) {
    float ap = TL_BIG, ng = TL_BIG; int cn = 0;
    for (int w = 0; w < 8; ++w) {
      ap = fminf(ap, s_ap_w[w][tid]);
      ng = fminf(ng, s_ng_w[w][tid]);
      cn += s_cnt_w[w][tid];
    }
    s_ap[tid] = ap; s_ng[tid] = ng; s_cnt[tid] = cn;
  }
  __syncthreads();

  // =========================== PASS 2 ===========================
  // first j (batch order) with: negative && dist_ap < d2 < dist_ap + margin
  float rap[8], rapM[8];
#pragma unroll
  for (int v = 0; v < 8; ++v) {
    rap[v]  = s_ap[half * 8 + v];
    rapM[v] = rap[v] + TL_MARGIN;
  }

  int fj[8];
#pragma unroll
  for (int v = 0; v < 8; ++v) fj[v] = TL_INTMAX;

  for (int t = wave; t < nColTiles; t += 8) {
    const int c = (t << 4) + l16;
    if (t + 8 < nColTiles)
      __builtin_prefetch(E + (size_t)(((t + 8) << 4) + l16) * TL_D, 0, 0);

    const float* brow = E + (size_t)c * TL_D + (half << 1);
    v2f bfrag[TL_NK];
#pragma unroll
    for (int k = 0; k < TL_NK; ++k)
      bfrag[k] = *(const v2f*)(brow + (k << 2));

    v8f acc0 = {}, acc1 = {};
#pragma unroll
    for (int k = 0; k < TL_NK; k += 2) {
      acc0 = __builtin_amdgcn_wmma_f32_16x16x4_f32(false, afrag[k],
                                                   false, bfrag[k],
                                                   (short)0, acc0, false, false);
      acc1 = __builtin_amdgcn_wmma_f32_16x16x4_f32(false, afrag[k + 1],
                                                   false, bfrag[k + 1],
                                                   (short)0, acc1, false, false);
    }
    const float csq  = sq[c];
    const int   clab = labels[c];

#pragma unroll
    for (int v = 0; v < 8; ++v) {
      const float g  = acc0[v] + acc1[v];
      const float d2 = fmaxf(rsq[v] + csq - 2.0f * g, 0.0f);
      const bool cond = (clab != rlab[v]) && (d2 > rap[v]) && (d2 < rapM[v]);
      const int  cand = cond ? c : TL_INTMAX;
      fj[v] = (cand < fj[v]) ? cand : fj[v];
    }
  }

#pragma unroll
  for (int v = 0; v < 8; ++v) {
    int f = fj[v];
    for (int m = 8; m >= 1; m >>= 1) {
      int o = __shfl_xor(f, m, 32);
      f = (o < f) ? o : f;
    }
    if (l16 == 0) s_fj_w[wave][half * 8 + v] = f;
  }
  __syncthreads();
  if (tid < 16) {
    int f = TL_INTMAX;
    for (int w = 0; w < 8; ++w) { int o = s_fj_w[w][tid]; f = (o < f) ? o : f; }
    s_fj[tid] = f;
  }
  __syncthreads();

  // =========================== PASS 3 ===========================
  // neg_rank ordinal: # negatives strictly before first_j (reference cumsum).
  {
    int fjr[16], labr[16], lcnt[16];
#pragma unroll
    for (int r = 0; r < 16; ++r) { fjr[r] = s_fj[r]; labr[r] = s_lab[r]; lcnt[r] = 0; }
    for (int j = tid; j < B; j += 256) {
      const int lj = labels[j];
#pragma unroll
      for (int r = 0; r < 16; ++r)
        lcnt[r] += ((j < fjr[r]) && (lj != labr[r])) ? 1 : 0;
    }
#pragma unroll
    for (int r = 0; r < 16; ++r)
      if (lcnt[r]) atomicAdd(&s_rank[r], lcnt[r]);
  }
  __syncthreads();

  // =========================== PASS 4 ===========================
  if (tid < 16) {
    const int r = tid;
    const float dist_ap = s_ap[r];
    float d_an;
    if (s_fj[r] != TL_INTMAX) {
      int rank = s_rank[r];
      rank = (rank > B - 1) ? (B - 1) : rank;          // reference clip
      const float* a = E + (size_t)(rowBase + r) * TL_D;
      const float* b = E + (size_t)rank * TL_D;
      float dot = 0.f;
      for (int k = 0; k < TL_D; ++k) dot = fmaf(a[k], b[k], dot);
      d_an = fmaxf(s_sqr[r] + sq[rank] - 2.0f * dot, 0.0f); // d2[i, rank]
    } else {
      d_an = s_ng[r];                                  // fallback: min over negatives
    }
    const bool valid = (s_cnt[r] >= 2);                // class has >= 2 members
    const float loss = fmaxf(dist_ap - d_an + TL_MARGIN, 0.0f);
    s_loss[r] = valid ? loss : 0.0f;
    s_val[r]  = valid ? 1.0f : 0.0f;
  }
  __syncthreads();
  if (tid == 0) {
    float ls = 0.f, vs = 0.f;
    for (int r = 0; r < 16; ++r) { ls += s_loss[r]; vs += s_val[r]; }
    partials[blockIdx.x * 2 + 0] = ls;
    partials[blockIdx.x * 2 + 1] = vs;
  }
}

__global__ void triplet_reduce_kernel(const float* __restrict__ partials,
                                      float* __restrict__ out, int nTiles) {
  __shared__ float s_l[256];
  __shared__ float s_c[256];
  float ls = 0.f, cs = 0.f;
  for (int i = threadIdx.x; i < nTiles; i += 256) {
    ls += partials[i * 2 + 0];
    cs += partials[i * 2 + 1];
  }
  s_l[threadIdx.x] = ls; s_c[threadIdx.x] = cs;
  __syncthreads();
  for (int s = 128; s > 0; s >>= 1) {
    if (threadIdx.x < s) {
      s_l[threadIdx.x] += s_l[threadIdx.x + s];
      s_c[threadIdx.x] += s_c[threadIdx.x + s];
    }
    __syncthreads();
  }
  if (threadIdx.x == 0)
    out[0] = (s_c[0] > 0.f) ? (s_l[0] / s_c[0]) : 0.f;
}

extern "C" void kernel_launch(void* const* d_in, const int* in_sizes, int n_in,
                              void* d_out, int out_size, void* d_ws, size_t ws_size,
                              hipStream_t stream) {
  (void)n_in; (void)out_size; (void)ws_size;
  const float* E      = (const float*)d_in[0];
  const int*   labels = (const int*)d_in[1];
  const int B = in_sizes[1];            // 8192 (D fixed at 64 by the reference)
  const int nRowTiles = B >> 4;         // 512 workgroups, 16 rows each

  float* sq       = (float*)d_ws;       // B floats, fully rewritten each call
  float* partials = sq + B;             // [nRowTiles][2], fully rewritten each call

  sqnorm_kernel<<<(B + 255) / 256, 256, 0, stream>>>(E, sq, B);
  triplet_tiles_kernel<<<nRowTiles, 256, 0, stream>>>(E, labels, sq, partials, B);
  triplet_reduce_kernel<<<1, 256, 0, stream>>>(partials, (float*)d_out, nRowTiles);
}